// Model_58506044506412
// MI455X (gfx1250) — compile-verified
//
#include <hip/hip_runtime.h>
#include <hip/hip_bf16.h>
#include <math.h>

typedef __attribute__((ext_vector_type(2))) float v2f;
typedef __attribute__((ext_vector_type(4))) float v4f;
typedef __attribute__((ext_vector_type(8))) float v8f;

// ---------------------------------------------------------------------------
// Kernel 1: conv1 (1->64, 5x5, VALID on 58x58 -> 54x54) fused with 2x2 maxpool
// -> h1 [256, 64, 27, 27] fp32 in workspace. ~2.4 GFLOP, VALU.
// ---------------------------------------------------------------------------
__global__ __launch_bounds__(256) void conv1_pool_kernel(
    const float* __restrict__ x, const float* __restrict__ w1,
    float* __restrict__ h1) {
  __shared__ float xs[58 * 58];  // 13.5 KB: whole input image
  const int b = blockIdx.x;
  const int tid = threadIdx.x;
  const float* xb = x + b * 58 * 58;
  for (int i = tid; i < 58 * 58; i += 256) xs[i] = xb[i];
  __syncthreads();

  for (int oc = 0; oc < 64; ++oc) {
    float w[25];
#pragma unroll
    for (int j = 0; j < 25; ++j) w[j] = w1[oc * 25 + j];  // uniform -> scalar loads
    for (int p = tid; p < 27 * 27; p += 256) {
      const int py = p / 27, px = p - (p / 27) * 27;
      const int iy = py * 2, ix = px * 2;
      float patch[6][6];
#pragma unroll
      for (int r = 0; r < 6; ++r)
#pragma unroll
        for (int c = 0; c < 6; ++c) patch[r][c] = xs[(iy + r) * 58 + ix + c];
      float m = -INFINITY;
#pragma unroll
      for (int dy = 0; dy < 2; ++dy)
#pragma unroll
        for (int dx = 0; dx < 2; ++dx) {
          float s = 0.f;
#pragma unroll
          for (int ky = 0; ky < 5; ++ky)
#pragma unroll
            for (int kx = 0; kx < 5; ++kx)
              s = fmaf(patch[dy + ky][dx + kx], w[ky * 5 + kx], s);
          m = fmaxf(m, s);
        }
      h1[((b * 64 + oc) * 27 + py) * 27 + px] = m;
    }
  }
}

// ---------------------------------------------------------------------------
// Kernel 2: conv2 (64->128, 5x5 VALID on 27x27 -> 23x23) as implicit GEMM on
// V_WMMA_F32_16X16X4_F32, with the 128->10 linear fused in the epilogue.
// M = 529 pixels (34 tiles of 16), N = 128 ch (8 tiles), K = 64 ci * 25 taps.
// Grid: (5 m-groups, 256 images); block = 256 threads = 8 waves; each wave
// owns one M-tile x all 8 N-tiles (8 x v8f accumulators).
//
// Weight chunk LDS layout (quad-interleaved for single ds_load_b128 per two
// N-tiles): element (k, n) with kp=k>>1, kb=k&1, nl=n&15, nt=n>>4 lives at
//   wlds[kp*320 + nl*20 + (nt>>1)*4 + (nt&1)*2 + kb]
// so a v4f at [kp][nl][ntp] = {B[k0,ntE], B[k1,ntE], B[k0,ntO], B[k1,ntO]},
// 16B-aligned, nl-stride 80B => conflict-free, and the two v2f WMMA operands
// are even-aligned sub-registers of the quad (no repacking movs).
// ---------------------------------------------------------------------------
__global__ __launch_bounds__(256) void conv2_wmma_linear_kernel(
    const float* __restrict__ h1, const float* __restrict__ w2,
    const float* __restrict__ lw, const float* __restrict__ lb,
    float* __restrict__ out) {
  __shared__ __attribute__((aligned(16))) float wlds[14 * 320]; // 17.9 KB
  __shared__ float alds[12 * 27 + 1];  // h1 rows + one always-zero slot [324]
  __shared__ float lwlds[10 * 132];    // linear weights, padded stride (banks)
  __shared__ float lblds[10];
  __shared__ float hlds[8 * 16 * 132]; // per-wave 16x128 conv2 tile, padded

  const int b = blockIdx.y;
  const int mgroup = blockIdx.x;       // 0..4
  const int tid = threadIdx.x;
  const int wid = tid >> 5;            // wave 0..7
  const int lane = tid & 31;
  const int nl = lane & 15;            // N (or local M) index within tile
  const int kh = lane >> 4;            // K-half select per ISA f32 layouts

  // One-time staging: zero wlds (so per-ci staging only writes the 3200 real
  // elements; the k=25..27 pad slots stay zero), linear weights, zero slot.
  for (int i = tid; i < 14 * 320; i += 256) wlds[i] = 0.f;
  for (int i = tid; i < 10 * 128; i += 256) {
    const int o = i >> 7, c = i & 127;
    lwlds[o * 132 + c] = lw[i];
  }
  if (tid < 10) lblds[tid] = lb[tid];
  if (tid == 0) alds[324] = 0.f;

  const int mtile = mgroup * 8 + wid;        // 0..39 (>=34 masked at store)
  const int m = mtile * 16 + nl;             // this lane's A-matrix row
  const int mc = m < 528 ? m : 528;          // clamp for address safety
  const int oy = mc / 23, ox = mc - (mc / 23) * 23;
  const int oy0 = (mgroup * 128) / 23;       // block-level staged row base
  const int abase = (oy - oy0) * 27 + ox;

  // ci-invariant A-fragment LDS offsets; padded taps route to the zero slot.
  int aoff0[7], aoff1[7];
#pragma unroll
  for (int j = 0; j < 7; ++j) {
    const int k0 = 4 * j + 2 * kh, k1 = k0 + 1;
    aoff0[j] = (k0 < 25) ? (k0 / 5) * 27 + (k0 % 5) : (324 - abase);
    aoff1[j] = (k1 < 25) ? (k1 / 5) * 27 + (k1 % 5) : (324 - abase);
  }

  v8f acc[8];
#pragma unroll
  for (int nt = 0; nt < 8; ++nt)
#pragma unroll
    for (int jv = 0; jv < 8; ++jv) acc[nt][jv] = 0.f;

  __syncthreads();  // wlds zero-fill / lwlds visible before first use

  for (int ci = 0; ci < 64; ++ci) {
    // Stage B chunk: k-fastest enumeration => coalesced w2 reads.
    for (int i = tid; i < 25 * 128; i += 256) {
      const int n = i / 25, k = i - n * 25;
      const int kp = k >> 1, kb = k & 1;
      const int nlw = n & 15, ntw = n >> 4;
      wlds[kp * 320 + nlw * 20 + (ntw >> 1) * 4 + (ntw & 1) * 2 + kb] =
          w2[(n * 64 + ci) * 25 + k];
    }
    // Stage A rows: h1[b, ci, oy0+r, c], zero-fill past row 26.
    for (int i = tid; i < 12 * 27; i += 256) {
      const int r = i / 27, c = i - r * 27;
      const int row = oy0 + r;
      alds[i] = (row < 27) ? h1[((b * 64 + ci) * 27 + row) * 27 + c] : 0.f;
    }
    if (ci + 1 < 64)  // pull next weight chunk toward L2 (global_prefetch_b8)
      __builtin_prefetch(&w2[((tid & 127) * 64 + ci + 1) * 25], 0, 1);
    __syncthreads();

#pragma unroll
    for (int j = 0; j < 7; ++j) {        // K = 28 in steps of 4
      // A fragment (16x4 f32 layout: lanes 0-15 K={k0,k0+1}, 16-31 shifted +2)
      v2f a;
      a.x = alds[abase + aoff0[j]];
      a.y = alds[abase + aoff1[j]];
      const v4f* wquad = (const v4f*)&wlds[(2 * j + kh) * 320 + nl * 20];
#pragma unroll
      for (int ntp = 0; ntp < 4; ++ntp) { // one b128 feeds two N-tiles
        const v4f q = wquad[ntp];
        v2f b0; b0.x = q.x; b0.y = q.y;
        v2f b1; b1.x = q.z; b1.y = q.w;
        acc[2 * ntp] = __builtin_amdgcn_wmma_f32_16x16x4_f32(
            false, a, false, b0, (short)0, acc[2 * ntp], false, false);
        acc[2 * ntp + 1] = __builtin_amdgcn_wmma_f32_16x16x4_f32(
            false, a, false, b1, (short)0, acc[2 * ntp + 1], false, false);
      }
    }
    __syncthreads();  // protect wlds/alds before next ci restage
  }

  // Epilogue: spill wave-private 16x128 tile to LDS (DS is in-order per wave),
  // then fused 128->10 linear + bias, masked store to [b,23,23,10].
  float* hw = &hlds[wid * 16 * 132];
#pragma unroll
  for (int nt = 0; nt < 8; ++nt)
#pragma unroll
    for (int jv = 0; jv < 8; ++jv)       // D layout: M = jv + 8*kh, N = nl
      hw[(jv + 8 * kh) * 132 + nt * 16 + nl] = acc[nt][jv];

#pragma unroll
  for (int i = 0; i < 5; ++i) {
    const int p = lane + 32 * i;         // 160 (pixel, out) pairs per wave
    const int pos = p / 10, o = p - pos * 10;
    float s = lblds[o];
#pragma unroll 8
    for (int c = 0; c < 128; ++c)
      s = fmaf(hw[pos * 132 + c], lwlds[o * 132 + c], s);
    const int mm = mtile * 16 + pos;
    if (mm < 529) {
      const int oyy = mm / 23, oxx = mm - (mm / 23) * 23;
      out[((b * 23 + oyy) * 23 + oxx) * 10 + o] = s;
    }
  }
}

// ---------------------------------------------------------------------------
extern "C" void kernel_launch(void* const* d_in, const int* in_sizes, int n_in,
                              void* d_out, int out_size, void* d_ws,
                              size_t ws_size, hipStream_t stream) {
  (void)in_sizes; (void)n_in; (void)out_size; (void)ws_size;
  const float* x  = (const float*)d_in[0];   // [256,1,58,58]
  const float* w1 = (const float*)d_in[1];   // [64,1,5,5]
  const float* w2 = (const float*)d_in[2];   // [128,64,5,5]
  const float* lw = (const float*)d_in[3];   // [10,128]
  const float* lb = (const float*)d_in[4];   // [10]
  float* out = (float*)d_out;                // [256,23,23,10]
  float* h1 = (float*)d_ws;                  // [256,64,27,27] = 47.8 MB scratch

  conv1_pool_kernel<<<256, 256, 0, stream>>>(x, w1, h1);
  conv2_wmma_linear_kernel<<<dim3(5, 256), 256, 0, stream>>>(h1, w2, lw, lb, out);
}